// AffinityPropagate_55241869361835
// MI455X (gfx1250) — compile-verified
//
#include <hip/hip_runtime.h>
#include <cstdint>

// Problem constants (match reference setup_inputs).
#define BATCH 8
#define K2    9
#define Hh    256
#define Ww    1216
#define HW    (Hh * Ww)          // 311296
#define TIMES 24

// Tiling: 64x16 output tile per 256-thread block; halo 66x18 in LDS.
#define TW    64
#define TH    16
#define LDSW  (TW + 2)           // 66
#define LDSH  (TH + 2)           // 18
#define LDSP  68                 // padded row stride (floats)

#ifndef __has_builtin
#define __has_builtin(x) 0
#endif

#if __has_builtin(__builtin_amdgcn_global_load_async_to_lds_b32)
#define HAVE_ASYNC_LDS 1
#else
#define HAVE_ASYNC_LDS 0
#endif

typedef __attribute__((address_space(1))) int gas_int;   // global
typedef __attribute__((address_space(3))) int las_int;   // LDS

__device__ __forceinline__ void async_copy_f32(const float* gsrc, float* ldst) {
#if HAVE_ASYNC_LDS
    // (global AS1 int* src, LDS AS3 int* dst, imm offset, imm cpol)
    __builtin_amdgcn_global_load_async_to_lds_b32(
        (gas_int*)(unsigned long long)(uintptr_t)gsrc,
        (las_int*)(uint32_t)(uintptr_t)ldst,
        0, 0);
#else
    *ldst = *gsrc;               // fallback: plain load + ds_store
#endif
}

__device__ __forceinline__ void wait_async_all() {
#if HAVE_ASYNC_LDS
#if __has_builtin(__builtin_amdgcn_s_wait_asynccnt)
    __builtin_amdgcn_s_wait_asynccnt(0);
#else
    asm volatile("s_wait_asynccnt 0" ::: "memory");
#endif
#endif
}

// One propagation step: dst = stencil(aff_normalized, src), zero-padded borders.
// aff layout [B,9,H,W]; src/dst layout [B,H,W] (C==1).
__global__ __launch_bounds__(256) void affinity_step(const float* __restrict__ aff,
                                                     const float* __restrict__ src,
                                                     float* __restrict__ dst) {
    __shared__ float lds[LDSH * LDSP];

    const int tid = threadIdx.x;
    const int x0  = blockIdx.x * TW;
    const int y0  = blockIdx.y * TH;
    const int b   = blockIdx.z;

    const float* fsrc = src + (size_t)b * HW;

    // ---- Stage (TH+2)x(TW+2) halo tile into LDS (async global->LDS) ----
    for (int i = tid; i < LDSH * LDSW; i += 256) {
        const int r  = i / LDSW;
        const int c  = i - r * LDSW;
        const int gy = y0 + r - 1;
        const int gx = x0 + c - 1;
        float* lp = &lds[r * LDSP + c];
        if ((unsigned)gy < (unsigned)Hh && (unsigned)gx < (unsigned)Ww) {
            async_copy_f32(&fsrc[(size_t)gy * Ww + gx], lp);
        } else {
            *lp = 0.0f;          // zero padding (ds_store; disjoint from async dests)
        }
    }
    wait_async_all();            // my wave's async writes to LDS are visible
    __syncthreads();             // everyone's writes visible (compiler adds ds wait)

    // ---- Compute 4 output pixels per thread ----
    const float* wbase = aff + (size_t)b * K2 * HW;
    float* fdst = dst + (size_t)b * HW;

#pragma unroll
    for (int k = 0; k < (TW * TH) / 256; ++k) {
        const int p  = k * 256 + tid;
        const int c  = p & (TW - 1);
        const int r  = p >> 6;           // TW == 64
        const int gy = y0 + r;
        const int gx = x0 + c;

        const float* wp = wbase + (size_t)gy * Ww + gx;
        float w[9];
#pragma unroll
        for (int n = 0; n < 9; ++n)
            w[n] = fabsf(wp[(size_t)n * HW]);   // 9 coalesced plane-strided loads (L2-resident)

        const float s = ((w[0] + w[1]) + (w[2] + w[3])) +
                        ((w[4] + w[5]) + (w[6] + w[7])) + w[8];

        float acc = 0.0f;
#pragma unroll
        for (int n = 0; n < 9; ++n) {
            const int di = n / 3;
            const int dj = n - 3 * di;
            acc = fmaf(w[n], lds[(r + di) * LDSP + (c + dj)], acc);
        }

        fdst[(size_t)gy * Ww + gx] = acc / s;
    }
}

extern "C" void kernel_launch(void* const* d_in, const int* in_sizes, int n_in,
                              void* d_out, int out_size, void* d_ws, size_t ws_size,
                              hipStream_t stream) {
    const float* aff  = (const float*)d_in[0];   // [8,9,256,1216] f32
    const float* feat = (const float*)d_in[1];   // [8,1,256,1216] f32
    float* out = (float*)d_out;                  // [8,1,256,1216] f32
    float* fws = (float*)d_ws;                   // one feature-sized ping buffer (~10 MB)

    const dim3 grid(Ww / TW, Hh / TH, BATCH);    // 19 x 16 x 8 blocks
    const dim3 block(256);

    // Ping-pong: even t writes ws, odd t writes d_out; TIMES=24 (even) => final in d_out.
    for (int t = 0; t < TIMES; ++t) {
        const float* src = (t == 0) ? feat : ((t & 1) ? fws : out);
        float* dst = (t & 1) ? out : fws;
        affinity_step<<<grid, block, 0, stream>>>(aff, src, dst);
    }
}